// SSIM_8083128451792
// MI455X (gfx1250) — compile-verified
//
#include <hip/hip_runtime.h>

// SSIM for [N=32, C=3, H=512, W=512] fp32 via separable 11-tap Gaussian blur
// expressed as banded-Toeplitz GEMMs on V_WMMA_F32_16X16X4_F32 (wave32).

typedef __attribute__((ext_vector_type(2))) float v2f;
typedef __attribute__((ext_vector_type(8))) float v8f;

#define IMG_H 512
#define IMG_W 512
#define TILE 16
#define HALO 26   // TILE + 11 - 1 input span
#define INR 32    // padded input rows: 2 x 16-row WMMA groups
#define INC 28    // padded input cols: 7 K-chunks x 4
#define KCHUNKS 7

// 11-tap Gaussian, sigma = 1.5, normalized (matches reference _gaussian_window 1D factor)
__device__ __constant__ float G11[11] = {
    0.00102838f, 0.00759876f, 0.03600077f, 0.10936070f, 0.21300553f,
    0.26601148f,
    0.21300553f, 0.10936070f, 0.03600077f, 0.00759876f, 0.00102838f};

// Banded Toeplitz entry: weight of input index c contributing to output index n.
__device__ __forceinline__ float wband(int c, int n) {
    int d = c - n;
    return (d >= 0 && d < 11) ? G11[d] : 0.0f;
}

__device__ __forceinline__ float qval(int q, float xv, float yv) {
    switch (q) {
        case 0:  return xv;
        case 1:  return yv;
        case 2:  return xv * xv;
        case 3:  return yv * yv;
        default: return xv * yv;
    }
}

__global__ void __launch_bounds__(32)
ssim_wmma_kernel(const float* __restrict__ x, const float* __restrict__ y,
                 float* __restrict__ out, float inv_count) {
    __shared__ float sx[INR][INC];
    __shared__ float sy[INR][INC];
    __shared__ float stmp[INR][TILE];

    const int lane  = threadIdx.x;            // 0..31 (wave32)
    const int tx    = blockIdx.x;
    const int ty    = blockIdx.y;
    const int plane = blockIdx.z;             // n*C + c

    const float* __restrict__ xp = x + (size_t)plane * IMG_H * IMG_W;
    const float* __restrict__ yp = y + (size_t)plane * IMG_H * IMG_W;

    const int oy0 = ty * TILE - 5;            // halo origin (pad = K/2 = 5)
    const int ox0 = tx * TILE - 5;

    // ---- stage zero-padded halo tiles into LDS ----
    for (int i = lane; i < INR * INC; i += 32) {
        int r = i / INC, c = i % INC;
        int gy = oy0 + r, gx = ox0 + c;
        float xv = 0.0f, yv = 0.0f;
        if (r < HALO && c < HALO && gy >= 0 && gy < IMG_H && gx >= 0 && gx < IMG_W) {
            size_t o = (size_t)gy * IMG_W + gx;
            xv = xp[o];
            yv = yp[o];
        }
        sx[r][c] = xv;
        sy[r][c] = yv;
    }
    __syncthreads();

    // ---- per-lane Toeplitz fragments (band is symmetric, so row-pass B
    //      fragment == col-pass A fragment; both use lane&15 and the K split) ----
    const int nn = lane & 15;                 // N for B/D, M for col-pass A
    const int kp = (lane < 16) ? 0 : 2;       // K pair base within a 4-chunk
    v2f wfrag[KCHUNKS];
    #pragma unroll
    for (int kc = 0; kc < KCHUNKS; ++kc) {
        int k0 = kc * 4 + kp;
        wfrag[kc].x = wband(k0 + 0, nn);
        wfrag[kc].y = wband(k0 + 1, nn);
    }

    const int mbase8 = (lane < 16) ? 0 : 8;   // D-fragment row base
    v8f res[5];

    #pragma unroll
    for (int q = 0; q < 5; ++q) {
        // ---- row pass: tmp[r][n] = sum_c in[r][c] * g[c-n], two 16-row groups ----
        #pragma unroll
        for (int g = 0; g < 2; ++g) {
            v8f acc = {};
            const int r = g * 16 + nn;        // A-fragment row (M = lane&15)
            #pragma unroll
            for (int kc = 0; kc < KCHUNKS; ++kc) {
                int c0 = kc * 4 + kp;
                v2f a;
                a.x = qval(q, sx[r][c0 + 0], sy[r][c0 + 0]);
                a.y = qval(q, sx[r][c0 + 1], sy[r][c0 + 1]);
                acc = __builtin_amdgcn_wmma_f32_16x16x4_f32(
                    false, a, false, wfrag[kc], (short)0, acc, false, false);
            }
            const int mb = g * 16 + mbase8;
            #pragma unroll
            for (int rr = 0; rr < 8; ++rr)
                stmp[mb + rr][nn] = acc[rr];
        }
        // ---- col pass: out[m][n] = sum_k g[k-m] * tmp[k][n] ----
        {
            v8f acc = {};
            #pragma unroll
            for (int kc = 0; kc < KCHUNKS; ++kc) {
                int k0 = kc * 4 + kp;
                v2f b;
                b.x = stmp[k0 + 0][nn];
                b.y = stmp[k0 + 1][nn];
                acc = __builtin_amdgcn_wmma_f32_16x16x4_f32(
                    false, wfrag[kc], false, b, (short)0, acc, false, false);
            }
            res[q] = acc;
        }
        __syncthreads();   // stmp reused next q (single-wave WG: DS stays in-order)
    }

    // ---- SSIM map on the 8 register-resident pixels per lane ----
    const float c1 = 1.0e-4f;   // 0.01^2
    const float c2 = 9.0e-4f;   // 0.03^2
    float s = 0.0f;
    #pragma unroll
    for (int r = 0; r < 8; ++r) {
        float mx = res[0][r], my = res[1][r];
        float mx2 = mx * mx, my2 = my * my, mxy = mx * my;
        float sx2 = res[2][r] - mx2;
        float sy2 = res[3][r] - my2;
        float sxy = res[4][r] - mxy;
        float num = (2.0f * mxy + c1) * (2.0f * sxy + c2);
        float den = (mx2 + my2 + c1) * (sx2 + sy2 + c2) + 1.0e-12f;
        s += num / den;
    }

    // ---- wave32 reduction + one atomic per tile ----
    #pragma unroll
    for (int off = 16; off > 0; off >>= 1)
        s += __shfl_xor(s, off, 32);
    if (lane == 0)
        atomicAdd(out, s * inv_count);
}

__global__ void zero_out_kernel(float* o) { o[0] = 0.0f; }

extern "C" void kernel_launch(void* const* d_in, const int* in_sizes, int n_in,
                              void* d_out, int out_size, void* d_ws, size_t ws_size,
                              hipStream_t stream) {
    const float* x = (const float*)d_in[0];
    const float* y = (const float*)d_in[1];
    float* out = (float*)d_out;

    const int planes = in_sizes[0] / (IMG_H * IMG_W);   // 32*3 = 96
    const float inv_count = 1.0f / ((float)planes * (float)IMG_H * (float)IMG_W);

    zero_out_kernel<<<1, 1, 0, stream>>>(out);

    dim3 grid(IMG_W / TILE, IMG_H / TILE, planes);      // 32 x 32 x 96 tiles
    ssim_wmma_kernel<<<grid, 32, 0, stream>>>(x, y, out, inv_count);
}